// Model_51453708206397
// MI455X (gfx1250) — compile-verified
//
#include <hip/hip_runtime.h>

// MLA KV-cache insert for MI455X (gfx1250), compile-only tuning.
// Bandwidth-bound (≈0.06 FLOP/B -> ~17us at 23.3 TB/s): no WMMA by design.
// CDNA5 paths used: global_load_async_to_lds_b128 (ASYNCcnt) staging,
// s_wait_asynccnt, wave32 xor-shuffle reduction, NT temporal hints on all
// streaming traffic (everything here is read-once/write-once except gamma).

typedef float __attribute__((ext_vector_type(4))) f32x4;

#define B_    4
#define S_    4096
#define L_    16384
#define RMS_  512
#define ROPE_ 64
#define H_    576

// LDS layout (float indices)
#define OFF_KV 0            // 576 floats: kv row (512 rms + 64 rope)
#define OFF_C  576          // 64 floats: cos
#define OFF_S2 640          // 64 floats: sin
#define OFF_G  704          // 512 floats: gamma
#define OFF_P  1216         // 4 floats: per-wave partial sums
#define LDS_FLOATS 1220

// CDNA5 ISA 10.2: for a generic pointer into the LDS aperture the aperture
// base lives entirely in addr[63:32]; the low 32 bits ARE the LDS byte
// offset that global_load_async_to_lds_* expects in its VDST VGPR.
__device__ __forceinline__ unsigned lds_off(const void* p) {
  return (unsigned)(unsigned long long)p;
}

// One-shot streaming data: NT so it doesn't displace gamma / hot lines in L2.
__device__ __forceinline__ void async_ld_b128_nt(unsigned lds_byte, const float* g) {
  asm volatile("global_load_async_to_lds_b128 %0, %1, off th:TH_LOAD_NT"
               :: "v"(lds_byte), "v"(g) : "memory");
}

// Re-read data (gamma: re-fetched by all 16384 blocks) -> default RT policy.
__device__ __forceinline__ void async_ld_b128(unsigned lds_byte, const float* g) {
  asm volatile("global_load_async_to_lds_b128 %0, %1, off"
               :: "v"(lds_byte), "v"(g) : "memory");
}

__device__ __forceinline__ void wait_async_all() {
  asm volatile("s_wait_asynccnt 0" ::: "memory");
}

// ---------------- main kernel: one 128-thread block (4 wave32) per token ---
__global__ __launch_bounds__(128) void mla_rope_rms_scatter(
    const float* __restrict__ kv, const float* __restrict__ gamma,
    const float* __restrict__ cosp, const float* __restrict__ sinp,
    const int* __restrict__ index,
    float* __restrict__ k_cache_out, float* __restrict__ ckv_cache_out,
    float* __restrict__ k_embed_out, float* __restrict__ v_out)
{
  __shared__ __align__(16) float sm[LDS_FLOATS];
  const int tok = blockIdx.x;         // tok = b*S + s
  const int b   = tok >> 12;          // S = 4096
  const int t   = threadIdx.x;        // 0..127

  const float* kv_row = kv + (size_t)tok * H_;

  // ---- async stage: kv[0..511] (round 1), gamma (round 2),
  //      kv[512..575] / cos / sin (round 3, 48 active lanes) ----
  async_ld_b128_nt(lds_off(&sm[OFF_KV + 4 * t]), kv_row + 4 * t);
  async_ld_b128(lds_off(&sm[OFF_G + 4 * t]), gamma + 4 * t);
  if (t < 48) {
    unsigned f; const float* g;
    if (t < 16)      { f = OFF_KV + 512 + 4 * t;  g = kv_row + 512 + 4 * t; }
    else if (t < 32) { f = OFF_C  + 4 * (t - 16); g = cosp + (size_t)tok * ROPE_ + 4 * (t - 16); }
    else             { f = OFF_S2 + 4 * (t - 32); g = sinp + (size_t)tok * ROPE_ + 4 * (t - 32); }
    async_ld_b128_nt(lds_off(&sm[f]), g);
  }
  wait_async_all();
  __syncthreads();

  // ---- RMS: sum of squares over 512 (4 per thread, wave32 xor-reduce) ----
  const f32x4 x = *(const f32x4*)&sm[OFF_KV + 4 * t];
  float ss = x.x * x.x + x.y * x.y + x.z * x.z + x.w * x.w;
#pragma unroll
  for (int m = 16; m >= 1; m >>= 1) ss += __shfl_xor(ss, m, 32);
  if ((t & 31) == 0) sm[OFF_P + (t >> 5)] = ss;
  __syncthreads();
  const float tot = sm[OFF_P + 0] + sm[OFF_P + 1] + sm[OFF_P + 2] + sm[OFF_P + 3];
  const float inv = rsqrtf(tot * (1.0f / (float)RMS_) + 1e-5f);

  const int idx = index[tok];  // scatter row within this batch's cache

  // ---- v = gamma * x * inv -> v_out + ckv_cache row (write-once -> NT) ----
  const f32x4 g4 = *(const f32x4*)&sm[OFF_G + 4 * t];
  f32x4 v4 = g4 * x * inv;
  __builtin_nontemporal_store(v4,
      (f32x4*)(v_out         + (size_t)tok * RMS_ + 4 * t));
  __builtin_nontemporal_store(v4,
      (f32x4*)(ckv_cache_out + ((size_t)b * L_ + (size_t)idx) * RMS_ + 4 * t));

  // ---- RoPE on the 64 rope features (threads 0..15, 4 each) ----
  // k[j]      = rope[2j]   (j<32),  k[32+m]   = rope[2m+1]
  // rot[j]    = -rope[2j+1](j<32),  rot[32+m] = rope[2m]
  if (t < 16) {
    const float* r = &sm[OFF_KV + RMS_];
    f32x4 ke;
#pragma unroll
    for (int u = 0; u < 4; ++u) {
      const int j = 4 * t + u;
      float kk, rot;
      if (j < 32) { kk = r[2 * j];           rot = -r[2 * j + 1]; }
      else        { const int m2 = j - 32; kk = r[2 * m2 + 1]; rot = r[2 * m2]; }
      ke[u] = kk * sm[OFF_C + j] + rot * sm[OFF_S2 + j];
    }
    __builtin_nontemporal_store(ke,
        (f32x4*)(k_embed_out + (size_t)tok * ROPE_ + 4 * t));
    __builtin_nontemporal_store(ke,
        (f32x4*)(k_cache_out + ((size_t)b * L_ + (size_t)idx) * ROPE_ + 4 * t));
  }
}

// ---------------- pass-through cache copy (302 MB R+W, fully streaming) ----
__global__ void copy_b128_nt(const f32x4* __restrict__ src,
                             f32x4* __restrict__ dst, long long n4) {
  long long i = (long long)blockIdx.x * blockDim.x + threadIdx.x;
  const long long stride = (long long)gridDim.x * blockDim.x;
  for (; i < n4; i += stride) {
    f32x4 v = __builtin_nontemporal_load(src + i);
    __builtin_nontemporal_store(v, dst + i);
  }
}

extern "C" void kernel_launch(void* const* d_in, const int* in_sizes, int n_in,
                              void* d_out, int out_size, void* d_ws, size_t ws_size,
                              hipStream_t stream) {
  const float* kv        = (const float*)d_in[0];
  const float* gamma     = (const float*)d_in[1];
  const float* cosp      = (const float*)d_in[2];
  const float* sinp      = (const float*)d_in[3];
  const int*   index     = (const int*)  d_in[4];
  const float* k_cache   = (const float*)d_in[5];
  const float* ckv_cache = (const float*)d_in[6];

  float* out           = (float*)d_out;
  float* k_cache_out   = out;                                        // B*L*ROPE
  float* ckv_cache_out = k_cache_out + (size_t)B_ * L_ * ROPE_;      // B*L*RMS
  float* k_embed_out   = ckv_cache_out + (size_t)B_ * L_ * RMS_;     // B*S*ROPE
  float* v_out         = k_embed_out + (size_t)B_ * S_ * ROPE_;      // B*S*RMS

  const long long nk4 = (long long)B_ * L_ * ROPE_ / 4;  // 1,048,576
  const long long nc4 = (long long)B_ * L_ * RMS_  / 4;  // 8,388,608

  copy_b128_nt<<<dim3(2048), dim3(256), 0, stream>>>(
      (const f32x4*)k_cache, (f32x4*)k_cache_out, nk4);
  copy_b128_nt<<<dim3(8192), dim3(256), 0, stream>>>(
      (const f32x4*)ckv_cache, (f32x4*)ckv_cache_out, nc4);

  mla_rope_rms_scatter<<<dim3(B_ * S_), dim3(128), 0, stream>>>(
      kv, gamma, cosp, sinp, index,
      k_cache_out, ckv_cache_out, k_embed_out, v_out);
}